// Superloss_22754736734408
// MI455X (gfx1250) — compile-verified
//
#include <hip/hip_runtime.h>
#include <math.h>

#define TPB   256
#define DEPTH 4       // async tiles in flight per group
#define NBLK  2048

#ifndef __has_builtin
#define __has_builtin(x) 0
#endif

#if __has_builtin(__builtin_amdgcn_global_load_async_to_lds_b128) && \
    __has_builtin(__builtin_amdgcn_s_wait_asynccnt)
#define USE_ASYNC_BUILTIN 1
#else
#define USE_ASYNC_BUILTIN 0
#warning "async-to-LDS builtins not found; using inline-asm path"
#endif

typedef int v4i_t __attribute__((ext_vector_type(4)));
typedef __attribute__((address_space(1))) v4i_t g_v4i;   // global int4
typedef __attribute__((address_space(3))) v4i_t l_v4i;   // LDS int4
typedef __attribute__((address_space(3))) void lds_void;

__device__ __forceinline__ void async_copy16(const float4* gp, float4* lp) {
#if USE_ASYNC_BUILTIN
  __builtin_amdgcn_global_load_async_to_lds_b128(
      (g_v4i*)(v4i_t*)gp,   // inner cast: drop const + bitcast; outer: addrspacecast
      (l_v4i*)(v4i_t*)lp,
      0, 0);
#else
  unsigned lds_off = (unsigned)(unsigned long long)(lds_void*)lp;
  asm volatile("global_load_async_to_lds_b128 %0, %1, off"
               :: "v"(lds_off), "v"((unsigned long long)gp)
               : "memory");
#endif
}

__device__ __forceinline__ void wait_async0() {
#if USE_ASYNC_BUILTIN
  __builtin_amdgcn_s_wait_asynccnt(0);
#else
  asm volatile("s_wait_asynccnt 0" ::: "memory");
#endif
}

// ---------------------------------------------------------------- constants
#define E_F      2.718281828459045f
#define GAMMA_F -0.7357588823428847f   // -2/e + 1e-12 (1e-12 vanishes in fp32)

// ---------------------------------------------------------------- W0 (Halley)
__device__ __forceinline__ float lambertw0f(float x) {
  // branchwise seed, identical to reference
  float xl  = fmaxf(x, 2.0f);
  float lg  = __logf(xl);
  float big = lg - __logf(lg);
  float p   = __fsqrt_rn(fmaxf(2.0f * (1.0f + E_F * x), 0.0f));
  float nearb  = -1.0f + p;
  float smallb = x * (1.0f - x);
  float w = (x > E_F) ? big : ((x < -0.25f) ? nearb : smallb);
  // Halley folded to one v_exp + one v_rcp per step:
  //   step = f / (ew*wp1 - (w+2)*f/(2*wp1)) = f*2*wp1 / (2*wp1^2*ew - (w+2)*f)
  // Cubic convergence from these seeds is fp32-exact by iter 2 on the clamped
  // domain; reference iters 5..8 are idempotent.
#pragma unroll
  for (int i = 0; i < 4; ++i) {
    float ew   = __expf(w);
    float f    = __fmaf_rn(w, ew, -x);
    float wp1  = w + 1.0f;
    float two1 = 2.0f * wp1;
    float num  = f * two1;
    float den  = __fmaf_rn(ew * two1, wp1, -(w + 2.0f) * f);
    w -= num * __builtin_amdgcn_rcpf(den);
  }
  return w;
}

__device__ __forceinline__ float super_elem(float l, float tau) {
  float beta  = l - tau;                       // LAM == 1
  float xx    = 0.5f * fmaxf(beta, GAMMA_F);
  float w     = lambertw0f(xx);
  float sigma = __expf(-w);
  // log(sigma) == -w exactly; LAM*log(sigma)^2 == w*w
  return __fmaf_rn(beta, sigma, w * w);
}

// ---------------------------------------------------------------- reductions
__device__ __forceinline__ float block_reduce_sum(float v) {
  __shared__ float sred[TPB / 32];
  const int lane = threadIdx.x & 31;
  const int wave = threadIdx.x >> 5;
#pragma unroll
  for (int o = 16; o > 0; o >>= 1) v += __shfl_down(v, o, 32);
  if (lane == 0) sred[wave] = v;
  __syncthreads();
  if (wave == 0) {
    v = (lane < (int)(blockDim.x >> 5)) ? sred[lane] : 0.0f;
#pragma unroll
    for (int o = 4; o > 0; o >>= 1) v += __shfl_down(v, o, 32);
  }
  return v;   // valid in thread 0
}

// ---------------------------------------------------------------- stage 1
// `full` = (n/4) / (gridDim.x*TPB), computed on host (avoids i64 div emulation)
__global__ void partial_sum_kernel(const float* __restrict__ x,
                                   float* __restrict__ partial,
                                   int n, int full) {
  __shared__ float4 tile[DEPTH][TPB];
  const int tid = threadIdx.x;
  const int S   = (int)gridDim.x * TPB;          // float4 stride
  const int g   = (int)blockIdx.x * TPB + tid;   // float4 index
  const int n4  = n >> 2;
  const float4* __restrict__ x4 = (const float4*)x;
  float acc = 0.0f;

  int it = 0;
  for (; it + DEPTH <= full; it += DEPTH) {
#pragma unroll
    for (int d = 0; d < DEPTH; ++d)
      async_copy16(&x4[g + (it + d) * S], &tile[d][tid]);
    wait_async0();
#pragma unroll
    for (int d = 0; d < DEPTH; ++d) {
      float4 v = tile[d][tid];
      acc += (v.x + v.y) + (v.z + v.w);
    }
  }
  for (; it < full; ++it) {
    float4 v = x4[g + it * S];
    acc += (v.x + v.y) + (v.z + v.w);
  }
  for (int i = full * S + g; i < n4; i += S) {   // ragged float4s
    float4 v = x4[i];
    acc += (v.x + v.y) + (v.z + v.w);
  }
  for (int i = (n4 << 2) + g; i < n; i += S) acc += x[i];   // scalar tail

  float bsum = block_reduce_sum(acc);
  if (tid == 0) partial[blockIdx.x] = bsum;
}

// ---------------------------------------------------------------- stage 3
__global__ void superloss_partial_kernel(const float* __restrict__ x,
                                         const float* __restrict__ tau_p,
                                         float* __restrict__ partial,
                                         int n, int full) {
  __shared__ float4 tile[DEPTH][TPB];
  const float tau = tau_p[0];
  const int tid = threadIdx.x;
  const int S   = (int)gridDim.x * TPB;
  const int g   = (int)blockIdx.x * TPB + tid;
  const int n4  = n >> 2;
  const float4* __restrict__ x4 = (const float4*)x;
  float acc = 0.0f;

  int it = 0;
  for (; it + DEPTH <= full; it += DEPTH) {
#pragma unroll
    for (int d = 0; d < DEPTH; ++d)
      async_copy16(&x4[g + (it + d) * S], &tile[d][tid]);
    wait_async0();
#pragma unroll
    for (int d = 0; d < DEPTH; ++d) {
      float4 v = tile[d][tid];
      acc += super_elem(v.x, tau) + super_elem(v.y, tau);
      acc += super_elem(v.z, tau) + super_elem(v.w, tau);
    }
  }
  for (; it < full; ++it) {
    float4 v = x4[g + it * S];
    acc += super_elem(v.x, tau) + super_elem(v.y, tau);
    acc += super_elem(v.z, tau) + super_elem(v.w, tau);
  }
  for (int i = full * S + g; i < n4; i += S) {
    float4 v = x4[i];
    acc += super_elem(v.x, tau) + super_elem(v.y, tau);
    acc += super_elem(v.z, tau) + super_elem(v.w, tau);
  }
  for (int i = (n4 << 2) + g; i < n; i += S) acc += super_elem(x[i], tau);

  float bsum = block_reduce_sum(acc);
  if (tid == 0) partial[blockIdx.x] = bsum;
}

// ---------------------------------------------------------------- stages 2 & 4
__global__ void reduce_finalize_kernel(const float* __restrict__ partial, int nb,
                                       float* __restrict__ out, float inv_n,
                                       int compute_tau) {
  float s = 0.0f;
  for (int i = threadIdx.x; i < nb; i += blockDim.x) s += partial[i];
  float t = block_reduce_sum(s);
  if (threadIdx.x == 0) {
    float mean = t * inv_n;
    if (compute_tau) {
      float tau = mean;                    // TAU0 == 0 -> tau = mean
      tau = 0.9f * tau + 0.1f * mean;      // FAC blend (== mean)
      out[0] = tau;
    } else {
      out[0] = mean;
    }
  }
}

// ---------------------------------------------------------------- launch
extern "C" void kernel_launch(void* const* d_in, const int* in_sizes, int n_in,
                              void* d_out, int out_size, void* d_ws, size_t ws_size,
                              hipStream_t stream) {
  const float* x = (const float*)d_in[0];
  const int n = in_sizes[0];

  int NB = NBLK;
  size_t need = (size_t)(2 * NB + 1) * sizeof(float);
  if (ws_size < need) {
    long long avail = (long long)(ws_size / sizeof(float));
    NB = (int)((avail - 1) / 2);
    if (NB < 1) NB = 1;
  }

  float* p1      = (float*)d_ws;    // [0, NB)       stage-1 partials
  float* tau_ptr = p1 + NB;         // [NB]          tau
  float* p3      = tau_ptr + 1;     // [NB+1, 2NB+1) stage-3 partials
  const float inv_n = 1.0f / (float)n;

  const int S    = NB * TPB;        // float4 stride per sweep
  const int full = (n >> 2) / S;    // uniform per-thread iterations

  partial_sum_kernel<<<NB, TPB, 0, stream>>>(x, p1, n, full);
  reduce_finalize_kernel<<<1, TPB, 0, stream>>>(p1, NB, tau_ptr, inv_n, 1);
  superloss_partial_kernel<<<NB, TPB, 0, stream>>>(x, tau_ptr, p3, n, full);
  reduce_finalize_kernel<<<1, TPB, 0, stream>>>(p3, NB, (float*)d_out, inv_n, 0);
}